// OptNetSudokuLearnA_3032246911380
// MI455X (gfx1250) — compile-verified
//
#include <hip/hip_runtime.h>

// ---------------------------------------------------------------------------
// Batched QP interior-point solver for MI455X (gfx1250, wave32).
//
// One wave32 = one batch element.  ~170 GFLOP total vs ~8 MB HBM traffic
// => purely compute bound; everything lives in LDS/registers.
// Schur complement S = A diag(h) A^T (padded 48x48, K=64) is computed with
// V_WMMA_F32_16X16X4_F32: 3x3 tiles x 16 K-steps = 144 WMMAs per iteration
// per wave, 9 independent v8f accumulators.
// ---------------------------------------------------------------------------

typedef float v2f __attribute__((ext_vector_type(2)));
typedef float v8f __attribute__((ext_vector_type(8)));

#define WPB        4            // waves (batch elems) per workgroup
#define BDIM       (WPB * 32)
#define YD         64           // y dimension
#define NEQ        40           // equality rows
#define NP         48           // NEQ padded to 3 WMMA tiles
#define AS_STRIDE  65           // 64+1: conflict-free row AND column access
#define S_STRIDE   49
#define N_ITERS    30
#define Q_PEN      0.1f
#define INV_MU_YD  (1.0f / 640.0f)   // 1/(MU * Y_DIM)

// Intra-wave LDS handoff: DS ops are in-order within a wave and lanes run in
// lockstep, so we only need to stop the compiler from reordering LDS traffic
// and drain DScnt before dependent cross-lane reads.
static __device__ __forceinline__ void wave_sync() {
  __builtin_amdgcn_wave_barrier();
  asm volatile("s_wait_dscnt 0" ::: "memory");
  __builtin_amdgcn_wave_barrier();
}

static __device__ __forceinline__ v8f wmma4(v2f a, v2f b, v8f c) {
  // 8-arg form: (neg_a, A, neg_b, B, c_mod, C, reuse_a, reuse_b)
  return __builtin_amdgcn_wmma_f32_16x16x4_f32(false, a, false, b,
                                               (short)0, c, false, false);
}

__global__ __launch_bounds__(BDIM) void
optnet_ipm_kernel(const float* __restrict__ puzzles,
                  const float* __restrict__ Ain,
                  float* __restrict__ out, int B)
{
  __shared__ float As[NP * AS_STRIDE];          // A, zero-padded rows 40..47
  __shared__ float Ssc[WPB][NP * S_STRIDE];     // per-wave Schur matrix
  __shared__ float vbuf[WPB][YD];               // per-wave 64-vector stage
  __shared__ float bbuf[WPB][NP];               // per-wave 48-vector stage
  __shared__ float hbuf[WPB][YD];               // per-wave H^{-1} diagonal

  const int tid  = threadIdx.x;
  const int wave = tid >> 5;
  const int lane = tid & 31;

  // Cooperative load of A into LDS (row-major, stride 65).
  for (int idx = tid; idx < NP * YD; idx += BDIM) {
    int r = idx >> 6, c = idx & 63;
    As[r * AS_STRIDE + c] = (r < NEQ) ? Ain[r * YD + c] : 0.0f;
  }
  __syncthreads();

  const int bi = blockIdx.x * WPB + wave;
  if (bi >= B) return;

  float* S  = Ssc[wave];
  float* vv = vbuf[wave];
  float* bb = bbuf[wave];
  float* hh = hbuf[wave];

  const int half = lane >> 4;     // WMMA lane-half (K sub-select)
  const int m    = lane & 15;     // WMMA row/col within tile

  // Linear term p = -flattened puzzle; lane owns elements {lane, lane+32}.
  const float p0 = -puzzles[bi * YD + lane];
  const float p1 = -puzzles[bi * YD + lane + 32];

  // State: z, lam (2/lane), nu (lane<32 -> nu[lane]; lane<8 -> nu[32+lane]).
  float z0 = 1.f, z1 = 1.f, l0 = 1.f, l1 = 1.f, nu0 = 0.f, nu1 = 0.f;

  for (int it = 0; it < N_ITERS; ++it) {
    // ---- Anu = A^T nu ----------------------------------------------------
    bb[lane] = nu0;
    if (lane < 16) bb[32 + lane] = (lane < 8) ? nu1 : 0.0f;
    wave_sync();
    float an0 = 0.f, an1 = 0.f;
    #pragma unroll 8
    for (int i = 0; i < NEQ; ++i) {
      float w = bb[i];
      an0 = __builtin_fmaf(w, As[i * AS_STRIDE + lane], an0);
      an1 = __builtin_fmaf(w, As[i * AS_STRIDE + lane + 32], an1);
    }
    const float rd0 = Q_PEN * z0 + p0 - l0 + an0;   // r_dual
    const float rd1 = Q_PEN * z1 + p1 - l1 + an1;

    // ---- r_pri = A z - 1 -------------------------------------------------
    vv[lane] = z0; vv[lane + 32] = z1;
    wave_sync();
    float rp0 = -1.0f, rp1 = -1.0f;
    {
      const float* Ar = &As[lane * AS_STRIDE];
      #pragma unroll 8
      for (int k = 0; k < YD; ++k) rp0 = __builtin_fmaf(Ar[k], vv[k], rp0);
      if (lane < 8) {
        const float* Ar2 = &As[(32 + lane) * AS_STRIDE];
        #pragma unroll 8
        for (int k = 0; k < YD; ++k) rp1 = __builtin_fmaf(Ar2[k], vv[k], rp1);
      }
    }

    // ---- duality gap, r1, H^{-1} -----------------------------------------
    float eta = l0 * z0 + l1 * z1;
    #pragma unroll
    for (int s = 16; s >= 1; s >>= 1) eta += __shfl_xor(eta, s, 32);
    const float inv_t = eta * INV_MU_YD;

    const float r10 = -rd0 - l0 + inv_t / z0;
    const float r11 = -rd1 - l1 + inv_t / z1;
    const float h0  = 1.0f / (Q_PEN + l0 / z0);
    const float h1  = 1.0f / (Q_PEN + l1 / z1);
    hh[lane] = h0; hh[lane + 32] = h1;
    wave_sync();

    // ---- S = A diag(h) A^T via V_WMMA_F32_16X16X4_F32 --------------------
    // A-op (16x4 f32): lane half selects K pair {0,1} / {2,3}.
    // B-op (4x16 f32): assumed mirror layout; B[K][N] = h[K] * A[N][K]
    // (diag(h) folded into the operand load -> no transpose build needed).
    v8f acc[3][3] = {};
    #pragma unroll
    for (int kk = 0; kk < 16; ++kk) {
      const int k0 = kk * 4 + half * 2;
      const float ha = hh[k0], hb2 = hh[k0 + 1];
      v2f aop[3], bop[3];
      #pragma unroll
      for (int t = 0; t < 3; ++t) {
        const float* row = &As[(t * 16 + m) * AS_STRIDE + k0];
        v2f a; a.x = row[0]; a.y = row[1];
        aop[t] = a;
        const int n = t * 16 + m;
        v2f bv;
        bv.x = ha  * As[n * AS_STRIDE + k0];
        bv.y = hb2 * As[n * AS_STRIDE + k0 + 1];
        bop[t] = bv;
      }
      #pragma unroll
      for (int ti = 0; ti < 3; ++ti)
        #pragma unroll
        for (int tk = 0; tk < 3; ++tk)
          acc[ti][tk] = wmma4(aop[ti], bop[tk], acc[ti][tk]);
    }
    // Spill S tiles to LDS (C/D layout: vgpr r, lane -> M = half*8+r, N = m).
    #pragma unroll
    for (int ti = 0; ti < 3; ++ti)
      #pragma unroll
      for (int tk = 0; tk < 3; ++tk)
        #pragma unroll
        for (int r = 0; r < 8; ++r)
          S[(ti * 16 + half * 8 + r) * S_STRIDE + tk * 16 + m] = acc[ti][tk][r];
    wave_sync();

    // ---- rhs = A (h * r1) + r_pri ----------------------------------------
    vv[lane] = h0 * r10; vv[lane + 32] = h1 * r11;
    wave_sync();
    float rhs0 = rp0, rhs1 = rp1;
    {
      const float* Ar = &As[lane * AS_STRIDE];
      #pragma unroll 8
      for (int k = 0; k < YD; ++k) rhs0 = __builtin_fmaf(Ar[k], vv[k], rhs0);
      if (lane < 8) {
        const float* Ar2 = &As[(32 + lane) * AS_STRIDE];
        #pragma unroll 8
        for (int k = 0; k < YD; ++k) rhs1 = __builtin_fmaf(Ar2[k], vv[k], rhs1);
      }
    }
    bb[lane] = rhs0;
    if (lane < 8) bb[32 + lane] = rhs1;
    wave_sync();

    // ---- LU factorization of S (40x40, SPD, no pivoting) -----------------
    for (int j = 0; j < NEQ - 1; ++j) {
      const float pinv = 1.0f / S[j * S_STRIDE + j];
      for (int r = j + 1 + lane; r < NEQ; r += 32) {
        const float f = S[r * S_STRIDE + j] * pinv;
        S[r * S_STRIDE + j] = f;
        for (int c = j + 1; c < NEQ; ++c)
          S[r * S_STRIDE + c] = __builtin_fmaf(-f, S[j * S_STRIDE + c],
                                               S[r * S_STRIDE + c]);
      }
      wave_sync();
    }
    // Forward substitution (unit lower L).
    for (int j = 0; j < NEQ; ++j) {
      const float yj = bb[j];
      for (int r = j + 1 + lane; r < NEQ; r += 32)
        bb[r] = __builtin_fmaf(-S[r * S_STRIDE + j], yj, bb[r]);
      wave_sync();
    }
    // Backward substitution (U).
    for (int j = NEQ - 1; j >= 0; --j) {
      const float xj = bb[j] / S[j * S_STRIDE + j];
      if (lane == 0) bb[j] = xj;
      for (int r = lane; r < j; r += 32)
        bb[r] = __builtin_fmaf(-S[r * S_STRIDE + j], xj, bb[r]);
      wave_sync();
    }
    const float dn0 = bb[lane];
    const float dn1 = (lane < 8) ? bb[32 + lane] : 0.0f;

    // ---- dz, dlam --------------------------------------------------------
    float da0 = 0.f, da1 = 0.f;               // dnu @ A = A^T dnu
    #pragma unroll 8
    for (int i = 0; i < NEQ; ++i) {
      float w = bb[i];
      da0 = __builtin_fmaf(w, As[i * AS_STRIDE + lane], da0);
      da1 = __builtin_fmaf(w, As[i * AS_STRIDE + lane + 32], da1);
    }
    const float dz0 = h0 * (r10 - da0);
    const float dz1 = h1 * (r11 - da1);
    const float dl0 = -l0 + inv_t / z0 - l0 * dz0 / z0;
    const float dl1 = -l1 + inv_t / z1 - l1 * dz1 / z1;

    // ---- fraction-to-boundary step length --------------------------------
    const float INF = 3.0e38f;
    float sm = INF;
    sm = fminf(sm, (dz0 < 0.f) ? -z0 / dz0 : INF);
    sm = fminf(sm, (dz1 < 0.f) ? -z1 / dz1 : INF);
    sm = fminf(sm, (dl0 < 0.f) ? -l0 / dl0 : INF);
    sm = fminf(sm, (dl1 < 0.f) ? -l1 / dl1 : INF);
    #pragma unroll
    for (int s = 16; s >= 1; s >>= 1) sm = fminf(sm, __shfl_xor(sm, s, 32));
    const float a = fminf(1.0f, 0.99f * sm);

    z0 += a * dz0; z1 += a * dz1;
    l0 += a * dl0; l1 += a * dl1;
    nu0 += a * dn0;
    if (lane < 8) nu1 += a * dn1;
    wave_sync();
  }

  out[bi * YD + lane]      = z0;
  out[bi * YD + lane + 32] = z1;
}

extern "C" void kernel_launch(void* const* d_in, const int* in_sizes, int n_in,
                              void* d_out, int out_size, void* d_ws, size_t ws_size,
                              hipStream_t stream) {
  const float* puzzles = (const float*)d_in[0];   // (16384, 4,4,4) f32
  const float* A       = (const float*)d_in[1];   // (40, 64) f32
  float* out = (float*)d_out;                     // (16384, 4,4,4) f32
  (void)d_ws; (void)ws_size; (void)n_in; (void)out_size;

  const int B = in_sizes[0] / YD;                 // batch = 16384
  const int grid = (B + WPB - 1) / WPB;           // 4096 workgroups
  optnet_ipm_kernel<<<dim3(grid), dim3(BDIM), 0, stream>>>(puzzles, A, out, B);
}